// SIMPLE_13950053778155
// MI455X (gfx1250) — compile-verified
//
#include <hip/hip_runtime.h>
#include <stdint.h>

// Problem dims (fixed by the reference): input [B,N,D] f32, mask [B,N] f32.
#define B_ 64
#define N_ 4096
#define D_ 128
#define NM (B_ * N_)          // 262144 row-elements
#define ROW_BYTES (D_ * 4)    // 512 bytes per row-element

// ---------------------------------------------------------------------------
// Phase 1: forward-fill index scan. One block (256 threads) per batch row.
// Each thread scans 16 contiguous positions; wave32 shfl max-scan; cross-wave
// prefix via LDS; wrap-around init = row's last observed index; clamp >= 0.
// ---------------------------------------------------------------------------
__global__ __launch_bounds__(256) void ff_scan_kernel(const float* __restrict__ mask,
                                                      int* __restrict__ gidx) {
  const int b    = blockIdx.x;
  const int t    = threadIdx.x;
  const int lane = t & 31;
  const int wid  = t >> 5;
  const int base = t * 16;

  const float* mrow = mask + (size_t)b * N_;

  int local[16];
  int run = -1;
#pragma unroll
  for (int i = 0; i < 16; ++i) {
    float m = mrow[base + i];
    int o = (m < 0.9f) ? (base + i) : -1;   // observed => own index, else -1
    run = (o > run) ? o : run;
    local[i] = run;                          // chunk-local inclusive running max
  }

  // Inclusive max-scan of chunk totals across the wave (wave32).
  int v = run;
#pragma unroll
  for (int off = 1; off < 32; off <<= 1) {
    int u = __shfl_up(v, off, 32);
    if (lane >= off) v = (v > u) ? v : u;
  }

  __shared__ int wpref[8];   // per-wave totals -> exclusive prefixes
  __shared__ int rowlast;    // row-wide max observed index
  if (lane == 31) wpref[wid] = v;
  __syncthreads();
  if (t == 0) {
    int acc = -1;
#pragma unroll
    for (int w = 0; w < 8; ++w) {
      int x = wpref[w];
      wpref[w] = acc;                        // exclusive prefix for wave w
      acc = (x > acc) ? x : acc;
    }
    rowlast = acc;
  }
  __syncthreads();

  const int wave_ex = wpref[wid];
  const int up      = __shfl_up(v, 1, 32);   // inclusive scan shifted by one lane
  const int th_ex   = (lane == 0) ? wave_ex : ((up > wave_ex) ? up : wave_ex);
  int rl = rowlast;
  if (rl < 0) rl = 0;                        // guard: row with no observed values

  int* orow = gidx + (size_t)b * N_ + base;
#pragma unroll
  for (int i = 0; i < 16; ++i) {
    int val = (local[i] > th_ex) ? local[i] : th_ex;   // inclusive cummax up to j
    if (val < 0) val = rl;                             // wrap-around init
    orow[i] = val;
  }
}

// ---------------------------------------------------------------------------
// Phase 2: gathered row copy via CDNA5 async global<->LDS DMA, SADDR form,
// two-buffer-set software pipeline (one s_wait_asynccnt per stage).
// 1024 blocks x 8 waves; each wave owns 32 consecutive row-elements (all in
// the same batch row since 4096 % 32 == 0), processed in batches of 8.
// Gather indices are read through the *scalar* path (readfirstlane'd base ->
// s_load), so per-op source addresses are SALU + one v_add.
// Lane l moves bytes [16l,16l+16) of each 512B row.
// LDS: 2 sets x 8 buffers x 512B = 8KB per wave, 64KB per block.
// ---------------------------------------------------------------------------
#define P2_BLOCKS 1024
#define WAVES_PER_BLOCK 8
#define TOTAL_WAVES (P2_BLOCKS * WAVES_PER_BLOCK)   // 8192
#define ELEMS_PER_WAVE (NM / TOTAL_WAVES)           // 32 (exact)
#define BATCH 8
#define NBATCH (ELEMS_PER_WAVE / BATCH)             // 4

__global__ __launch_bounds__(256) void gather_copy_async_kernel(
    const float* __restrict__ input, const int* __restrict__ gidx,
    float* __restrict__ out) {
  __shared__ float lds[WAVES_PER_BLOCK * 2 * BATCH * (ROW_BYTES / 4)];  // 64 KB

  const int t     = threadIdx.x;
  const int lane  = t & 31;
  const int wid   = t >> 5;
  const int gwave = blockIdx.x * WAVES_PER_BLOCK + wid;
  const int e0    = gwave * ELEMS_PER_WAVE;
  if (e0 >= NM) return;

  // Wave-uniform base: readfirstlane makes uniformity provable so the 32
  // gather indices come in via scalar loads (s_load, KMcnt) and stay in SGPRs.
  const int e0u = __builtin_amdgcn_readfirstlane(e0);
  const int* __restrict__ gu = gidx + e0u;

  const int b = e0u >> 12;                      // N_ == 4096; constant per wave
  const unsigned brow = (unsigned)(b << 12);    // first row-element of batch b

  // LDS byte offsets of this wave's two staging sets (low 32 bits of the flat
  // shared-memory address are the DS-relative byte offset on CDNA5).
  const unsigned lds_wave =
      (unsigned)(uintptr_t)(&lds[wid * 2 * BATCH * (ROW_BYTES / 4)]);
  const unsigned lane_off = (unsigned)(lane * 16);
  const unsigned setoff[2] = {lds_wave + lane_off,
                              lds_wave + (unsigned)(BATCH * ROW_BYTES) + lane_off};

  // dst byte offset of this wave's first element
  const unsigned dst0 = (unsigned)e0u * (unsigned)ROW_BYTES + lane_off;

  // ---- batch issue helpers (fully unrolled; g is SGPR-resident) ----
#define LOAD_BATCH(bt, setbase)                                                \
  _Pragma("unroll") for (int k = 0; k < BATCH; ++k) {                          \
    const unsigned g = (unsigned)gu[(bt) * BATCH + k];                         \
    const unsigned soff = (brow + g) * (unsigned)ROW_BYTES + lane_off;         \
    const unsigned laddr = (setbase) + (unsigned)(k * ROW_BYTES);              \
    asm volatile("global_load_async_to_lds_b128 %0, %1, %2"                    \
                 :: "v"(laddr), "v"(soff), "s"(input) : "memory");             \
  }

#define STORE_BATCH(bt, setbase)                                               \
  _Pragma("unroll") for (int k = 0; k < BATCH; ++k) {                          \
    const unsigned doff = dst0 + (unsigned)(((bt) * BATCH + k) * ROW_BYTES);   \
    const unsigned laddr = (setbase) + (unsigned)(k * ROW_BYTES);              \
    asm volatile("global_store_async_from_lds_b128 %0, %1, %2 th:TH_STORE_NT"  \
                 :: "v"(doff), "v"(laddr), "s"(out) : "memory");               \
  }

#define WAIT_ASYNC0() asm volatile("s_wait_asynccnt 0x0" ::: "memory")

  // ---- software pipeline: load(i+1) overlaps store(i) ----
  LOAD_BATCH(0, setoff[0]);
  WAIT_ASYNC0();
#pragma unroll
  for (int bt = 0; bt < NBATCH; ++bt) {
    const unsigned cur = setoff[bt & 1];
    STORE_BATCH(bt, cur);
    if (bt + 1 < NBATCH) {
      const unsigned nxt = setoff[(bt + 1) & 1];
      LOAD_BATCH(bt + 1, nxt);
    }
    WAIT_ASYNC0();   // drains store(bt) and (if issued) load(bt+1)
  }

#undef LOAD_BATCH
#undef STORE_BATCH
#undef WAIT_ASYNC0
}

// ---------------------------------------------------------------------------
extern "C" void kernel_launch(void* const* d_in, const int* in_sizes, int n_in,
                              void* d_out, int out_size, void* d_ws, size_t ws_size,
                              hipStream_t stream) {
  (void)in_sizes; (void)n_in; (void)out_size; (void)ws_size;

  const float* input = (const float*)d_in[0];   // [B,N,D] f32
  const float* mask  = (const float*)d_in[1];   // [B,N]   f32
  float*       out   = (float*)d_out;           // [B,N,D] f32
  int*         gidx  = (int*)d_ws;              // [B,N] gather indices (1 MiB)

  ff_scan_kernel<<<B_, 256, 0, stream>>>(mask, gidx);
  gather_copy_async_kernel<<<P2_BLOCKS, 256, 0, stream>>>(input, gidx, out);
}